// MDRNN_3805341024620
// MI455X (gfx1250) — compile-verified
//
#include <hip/hip_runtime.h>
#include <cmath>

// Problem dims (match reference)
#define B_ 32
#define H_ 64
#define W_ 64
#define D_ 256
#define U_ 512

typedef __attribute__((ext_vector_type(2))) float v2f;
typedef __attribute__((ext_vector_type(8))) float v8f;

// ---- fragment loaders for V_WMMA_F32_16X16X4_F32 ---------------------------
// A (16x4, MxK) layout: lanes 0-15 hold rows M=0..15 with {K=k0,k0+1} in the
// two VGPRs; lanes 16-31 hold the same rows with {K=k0+2,k0+3}.
__device__ __forceinline__ v2f load_a_frag(const float* __restrict__ base,
                                           long rowStride, int k0, int lane) {
    const int r  = lane & 15;
    const int kk = k0 + ((lane >> 4) << 1);
    const float* p = base + (long)r * rowStride + kk;
    v2f a = { p[0], p[1] };
    return a;
}

// B (4x16, KxN) layout: lane n%16 selects column; lanes 0-15 hold K=k0 (v0)
// and K=k0+1 (v1); lanes 16-31 hold K=k0+2 / k0+3.
__device__ __forceinline__ v2f load_b_frag(const float* __restrict__ Bm, int ldb,
                                           int n0, int k0, int lane) {
    const int n  = n0 + (lane & 15);
    const int kk = k0 + ((lane >> 4) << 1);
    v2f b = { Bm[(long)kk * ldb + n], Bm[(long)(kk + 1) * ldb + n] };
    return b;
}

// Accumulate a (32 x K) * (K x 64) slab into 2x4 f32 16x16 tiles via WMMA.
__device__ __forceinline__ void gemm_acc(const float* __restrict__ Abase, long aRowStride,
                                         const float* __restrict__ Bm, int ldb, int K,
                                         int nBase, int lane, v8f acc[2][4]) {
    #pragma unroll 4
    for (int k0 = 0; k0 < K; k0 += 4) {
        v2f a0 = load_a_frag(Abase,                   aRowStride, k0, lane); // rows  0..15
        v2f a1 = load_a_frag(Abase + 16 * aRowStride, aRowStride, k0, lane); // rows 16..31
        #pragma unroll
        for (int t = 0; t < 4; ++t) {
            v2f b = load_b_frag(Bm, ldb, nBase + 16 * t, k0, lane);
            acc[0][t] = __builtin_amdgcn_wmma_f32_16x16x4_f32(
                false, a0, false, b, (short)0, acc[0][t], false, false);
            acc[1][t] = __builtin_amdgcn_wmma_f32_16x16x4_f32(
                false, a1, false, b, (short)0, acc[1][t], false, false);
        }
    }
}

// One anti-diagonal step. Block = one grid cell (i,j); 8 wave32 = 256 threads.
// h[i,j] = tanh(x[i,j]@wax + h[i-1,j]@w0 + h[i,j-1]@w1 + ba)
__global__ void __launch_bounds__(256)
mdrnn_diag_kernel(const float* __restrict__ x,   const float* __restrict__ wax,
                  const float* __restrict__ w0,  const float* __restrict__ w1,
                  const float* __restrict__ ba,
                  const float* __restrict__ hprev,  // diag d-1: [H][B][U]
                  float* __restrict__ hcur,         // diag d:   [H][B][U]
                  float* __restrict__ out,          // (B,U)
                  int d, int i0)
{
    const int i = i0 + (int)blockIdx.x;
    const int j = d - i;                     // valid by construction of the launch
    const int tid   = (int)threadIdx.x;
    const int lane  = tid & 31;
    const int nBase = (tid >> 5) * 64;       // each wave owns 64 output columns

    v8f acc[2][4];
    #pragma unroll
    for (int mt = 0; mt < 2; ++mt)
        #pragma unroll
        for (int t = 0; t < 4; ++t)
            acc[mt][t] = (v8f){0.f, 0.f, 0.f, 0.f, 0.f, 0.f, 0.f, 0.f};

    // 1) input projection: rows of A are batch b, row stride H*W*D
    {
        const long xRowStride = (long)H_ * W_ * D_;
        const float* xBase = x + ((long)i * W_ + j) * (long)D_;
        gemm_acc(xBase, xRowStride, wax, U_, D_, nBase, lane, acc);
    }
    // 2) top neighbor h[i-1,j] @ w0   (block-uniform guard -> EXEC stays full)
    if (i > 0)
        gemm_acc(hprev + (long)(i - 1) * B_ * U_, U_, w0, U_, U_, nBase, lane, acc);
    // 3) left neighbor h[i,j-1] @ w1
    if (j > 0)
        gemm_acc(hprev + (long)i * B_ * U_,       U_, w1, U_, U_, nBase, lane, acc);

    // bias + tanh + store. C/D layout: VGPR r -> row r (lanes 0-15) / r+8 (16-31)
    const bool last  = (i == H_ - 1) && (j == W_ - 1);
    const int  n     = lane & 15;
    const int  hiRow = (lane >> 4) << 3;
    #pragma unroll
    for (int mt = 0; mt < 2; ++mt) {
        #pragma unroll
        for (int t = 0; t < 4; ++t) {
            const int col = nBase + 16 * t + n;
            const float bv = ba[col];
            #pragma unroll
            for (int r = 0; r < 8; ++r) {
                const int row = mt * 16 + hiRow + r;
                const float v = tanhf(acc[mt][t][r] + bv);
                hcur[((long)i * B_ + row) * U_ + col] = v;
                if (last) out[(long)row * U_ + col] = v;
            }
        }
    }
}

extern "C" void kernel_launch(void* const* d_in, const int* in_sizes, int n_in,
                              void* d_out, int out_size, void* d_ws, size_t ws_size,
                              hipStream_t stream) {
    (void)in_sizes; (void)n_in; (void)out_size; (void)ws_size;
    const float* x   = (const float*)d_in[0];
    const float* wax = (const float*)d_in[1];
    const float* w0  = (const float*)d_in[2];
    const float* w1  = (const float*)d_in[3];
    const float* ba  = (const float*)d_in[4];
    float* out  = (float*)d_out;
    float* hbuf = (float*)d_ws;                       // 2 * H*B*U f32 = 8 MB
    const long diagElems = (long)H_ * B_ * U_;

    // Wavefront over anti-diagonals; ping-pong diagonal state in workspace.
    for (int d = 0; d < H_ + W_ - 1; ++d) {
        const int i0 = (d > W_ - 1) ? d - (W_ - 1) : 0;
        const int i1 = (d < H_ - 1) ? d : (H_ - 1);
        const int nb = i1 - i0 + 1;                   // valid cells on this diagonal
        float*       hc = hbuf + (long)(d & 1) * diagElems;
        const float* hp = hbuf + (long)((d & 1) ^ 1) * diagElems;
        mdrnn_diag_kernel<<<dim3(nb), dim3(256), 0, stream>>>(
            x, wax, w0, w1, ba, hp, hc, out, d, i0);
    }
}